// SIR_25855703122403
// MI455X (gfx1250) — compile-verified
//
#include <hip/hip_runtime.h>
#include <stdint.h>

#define WG 256

// ---- TDM availability / arity detection -----------------------------------
#if defined(__has_builtin)
#if __has_builtin(__builtin_amdgcn_tensor_load_to_lds)
#define HAVE_TDM 1
#endif
#endif

#if defined(__has_include)
#if __has_include(<hip/amd_detail/amd_gfx1250_TDM.h>)
#define TDM_6ARG 1
#endif
#endif
#ifndef TDM_6ARG
#define TDM_6ARG 0
#endif

typedef unsigned int v4u __attribute__((ext_vector_type(4)));
typedef int          v8i __attribute__((ext_vector_type(8)));
typedef int          v4i __attribute__((ext_vector_type(4)));

// ---------------------------------------------------------------------------
// Phase 1: serial checkpoint walk with exact (bitwise) fixed-point exit.
// Unroll 4 steps per check; on a 4-step-snapshot match, verify single-step
// idempotence non-destructively (F(s)==s bitwise => fixed forever in fp32).
// ws layout:  int   ws[0]    = fp_seg (first segment whose checkpoint is the
//                                       fixed state; == nseg if none)
//             float ws[1..3] = fixed/final state {S,I,R}
//             float ws[16..] = checkpoints, 3 floats per segment
// ---------------------------------------------------------------------------
__global__ __launch_bounds__(32) void sir_phase1(
    const float* __restrict__ x, const float* __restrict__ bw,
    const float* __restrict__ gw, float* __restrict__ ws,
    long long steps, int C, int nseg)
{
    if (threadIdx.x != 0) return;

    float S = x[0], I = x[1], R = x[2];
    float pop     = S + I + R;              // population = x.sum()
    float inv_pop = 1.0f / pop;
    float b = bw[0], g = gw[0];

    float* ck = ws + 16;
    ck[0] = S; ck[1] = I; ck[2] = R;        // checkpoint for segment 0 = x

    int  fp_seg = nseg;
    bool fixed  = false;
    for (int seg = 1; seg < nseg && !fixed; ++seg) {
        for (int j = 0; j < C; j += 4) {    // C is a multiple of 4
            float oS = S, oI = I, oR = R;
#pragma unroll
            for (int u = 0; u < 4; ++u) {
                float P   = S * I;          // critical path: mul + 2 fma
                float inf = b * P * inv_pop;
                float rec = g * I;
                S = S - inf;
                I = I + inf - rec;
                R = R + rec;
            }
            if (__float_as_uint(S) == __float_as_uint(oS) &&
                __float_as_uint(I) == __float_as_uint(oI) &&
                __float_as_uint(R) == __float_as_uint(oR)) {
                // F^4(s)==s; confirm F(s)==s without committing state
                float P   = S * I;
                float inf = b * P * inv_pop;
                float rec = g * I;
                float nS  = S - inf;
                float nI  = I + inf - rec;
                float nR  = R + rec;
                if (__float_as_uint(nS) == __float_as_uint(S) &&
                    __float_as_uint(nI) == __float_as_uint(I) &&
                    __float_as_uint(nR) == __float_as_uint(R)) {
                    fixed = true;
                    break;
                }
            }
        }
        if (fixed) { fp_seg = seg; break; }
        ck[seg * 3 + 0] = S; ck[seg * 3 + 1] = I; ck[seg * 3 + 2] = R;
    }
    ((int*)ws)[0] = fp_seg;
    ws[1] = S; ws[2] = I; ws[3] = R;
}

// ---------------------------------------------------------------------------
// Phase 2: one thread per segment of C rows.
//   - Workgroup stages its (up to 256) compute-path checkpoints into LDS with
//     ONE Tensor-Data-Mover load (TENSORcnt) issued by wave 0, then barriers.
//   - seg <  fp_seg : recompute C steps, register-pack 4 rows -> 3 b128 stores.
//   - seg >= fp_seg : constant fill via CDNA5 async LDS->global b128 stores
//     (ASYNCcnt): a 48B LDS pattern serves every 48B output chunk because the
//     instruction offset applies to both the LDS and global address.
// ---------------------------------------------------------------------------
__global__ __launch_bounds__(WG) void sir_phase2(
    const float* __restrict__ ws, const float* __restrict__ x,
    const float* __restrict__ bw, const float* __restrict__ gw,
    float* __restrict__ out, long long steps, int C, int nseg)
{
    __shared__ __align__(16) float ckl[WG * 3];   // TDM-staged checkpoints
    __shared__ __align__(16) float pat[WG * 12];  // 48B async-store pattern/lane

    int       tid     = threadIdx.x;
    long long wgSegLo = (long long)blockIdx.x * WG;
    int       seg     = (int)wgSegLo + tid;

    int fp_seg = ((const int*)ws)[0];

    // # segments in this WG that take the compute path (need checkpoints)
    long long lim = (long long)(fp_seg < nseg ? fp_seg : nseg);
    long long nl  = lim - wgSegLo;
    int nload = nl < 0 ? 0 : (nl > WG ? WG : (int)nl);

#ifdef HAVE_TDM
    if (nload > 0 && tid == 0) {
        unsigned N   = (unsigned)nload * 3u;                       // dwords
        unsigned long long ga =
            (unsigned long long)(const void*)(ws + 16 + wgSegLo * 3);
        unsigned lds = (unsigned)(uintptr_t)(&ckl[0]);             // WG-relative

        // D# group 0: count=1 | lds_addr | global_addr[56:0] | type=2
        v4u g0 = { 1u,
                   lds,
                   (unsigned)(ga & 0xFFFFFFFFu),
                   (unsigned)((ga >> 32) & 0x01FFFFFFu) | (2u << 30) };
        // D# group 1: data_size=4B; 1-D tensor/tile of N dwords
        v8i g1 = { (int)(2u << 16),                     // data_size=2 (4B)
                   (int)(N << 16),                      // tensor_dim0[15:0]
                   (int)((N >> 16) | (1u << 16)),       // dim0 hi | tensor_dim1=1
                   (int)(N << 16),                      // tile_dim0 = N
                   1,                                   // tile_dim1 = 1
                   (int)N,                              // tensor_dim0_stride
                   0, 0 };
        v4i g2 = { 0, 0, 0, 0 };
        v4i g3 = { 0, 0, 0, 0 };
#if TDM_6ARG
        v8i gz = { 0, 0, 0, 0, 0, 0, 0, 0 };
        __builtin_amdgcn_tensor_load_to_lds(g0, g1, g2, g3, gz, 0);
#else
        __builtin_amdgcn_tensor_load_to_lds(g0, g1, g2, g3, 0);
#endif
        __builtin_amdgcn_s_wait_tensorcnt(0);           // tile resident in LDS
    }
#else
    if (tid < nload) {
        ckl[tid * 3 + 0] = ws[16 + (wgSegLo + tid) * 3 + 0];
        ckl[tid * 3 + 1] = ws[16 + (wgSegLo + tid) * 3 + 1];
        ckl[tid * 3 + 2] = ws[16 + (wgSegLo + tid) * 3 + 2];
    }
#endif
    __syncthreads();                                    // publish LDS block

    if (seg >= nseg) return;

    long long t0    = (long long)seg * (long long)C;
    long long remll = steps - t0;
    int       cnt   = remll < (long long)C ? (int)remll : C;
    float*    o     = out + t0 * 3;

    if (seg >= fp_seg) {
        // ---- constant-fill fast path (bulk of the 48 MB output) ----
        float S = ws[1], I = ws[2], R = ws[3];
        int li = tid * 12;
        pat[li + 0] = S; pat[li + 1]  = I; pat[li + 2]  = R;
        pat[li + 3] = S; pat[li + 4]  = I; pat[li + 5]  = R;
        pat[li + 6] = S; pat[li + 7]  = I; pat[li + 8]  = R;
        pat[li + 9] = S; pat[li + 10] = I; pat[li + 11] = R;

        unsigned lds0 = (unsigned)(uintptr_t)(&pat[li]); // LDS byte address
        asm volatile("s_wait_dscnt 0x0" ::: "memory");   // pattern visible

        int   full = cnt >> 2;                           // 4 rows = 48 B
        char* gp   = (char*)o;
        for (int k = 0; k < full; ++k) {
            asm volatile(
                "global_store_async_from_lds_b128 %0, %1, off\n\t"
                "global_store_async_from_lds_b128 %0, %1, off offset:16\n\t"
                "global_store_async_from_lds_b128 %0, %1, off offset:32"
                :: "v"(gp), "v"(lds0) : "memory");
            gp += 48;
        }
        for (int j = full * 4; j < cnt; ++j) {           // tail rows
            o[j * 3 + 0] = S; o[j * 3 + 1] = I; o[j * 3 + 2] = R;
        }
        asm volatile("s_wait_asynccnt 0x0" ::: "memory");
        return;
    }

    // ---- recompute path (segments before the fixed point) ----
    float S = ckl[tid * 3 + 0];
    float I = ckl[tid * 3 + 1];
    float R = ckl[tid * 3 + 2];
    float pop     = x[0] + x[1] + x[2];
    float inv_pop = 1.0f / pop;
    float b = bw[0], g = gw[0];

    int j = 0;
    for (; j + 4 <= cnt; j += 4) {
        float r[12];
#pragma unroll
        for (int u = 0; u < 4; ++u) {
            r[u * 3 + 0] = S; r[u * 3 + 1] = I; r[u * 3 + 2] = R;
            float P   = S * I;
            float inf = b * P * inv_pop;
            float rec = g * I;
            S = S - inf;
            I = I + inf - rec;
            R = R + rec;
        }
        char* p = (char*)o + (size_t)j * 12;   // 48B chunk, 16B-aligned
        *(float4*)(p)      = make_float4(r[0], r[1], r[2],  r[3]);
        *(float4*)(p + 16) = make_float4(r[4], r[5], r[6],  r[7]);
        *(float4*)(p + 32) = make_float4(r[8], r[9], r[10], r[11]);
    }
    for (; j < cnt; ++j) {                     // tail (last segment only)
        o[j * 3 + 0] = S; o[j * 3 + 1] = I; o[j * 3 + 2] = R;
        float P   = S * I;
        float inf = b * P * inv_pop;
        float rec = g * I;
        S = S - inf;
        I = I + inf - rec;
        R = R + rec;
    }
}

// ---------------------------------------------------------------------------
extern "C" void kernel_launch(void* const* d_in, const int* in_sizes, int n_in,
                              void* d_out, int out_size, void* d_ws, size_t ws_size,
                              hipStream_t stream)
{
    const float* x  = (const float*)d_in[0];
    const float* bw = (const float*)d_in[1];
    const float* gw = (const float*)d_in[2];
    float*       out = (float*)d_out;

    long long steps = (long long)out_size / 3;   // out is (steps, 3)
    if (steps <= 0) return;

    // Segment length: 64 rows (multiple of 4 => 48B-chunk alignment holds).
    // Grow C if the checkpoint array would not fit in the workspace.
    int C = 64;
    long long nseg = (steps + C - 1) / C;
    while ((size_t)(64 + nseg * 12) > ws_size && C < (1 << 28)) {
        C <<= 1;
        nseg = (steps + C - 1) / C;
    }

    sir_phase1<<<1, 32, 0, stream>>>(x, bw, gw, (float*)d_ws, steps, C, (int)nseg);

    int blocks = (int)((nseg + WG - 1) / WG);
    if (blocks > 0) {
        sir_phase2<<<blocks, WG, 0, stream>>>((const float*)d_ws, x, bw, gw,
                                              out, steps, C, (int)nseg);
    }
}